// DTMLoss_11390253269371
// MI455X (gfx1250) — compile-verified
//
#include <hip/hip_runtime.h>

// Problem constants
#define N 8192
#define D 256
#define PANEL 64                 // columns staged per LDS panel
#define NPANELS (N / PANEL)      // 128
#define ROWS_PER_BLOCK 128       // 8 waves x 16 rows
#define LDSB_STRIDE 260          // 256 + 4 pad floats -> conflict-free b64 reads
#define KTOP 5                   // K+1 smallest (includes self-distance 0)

typedef __attribute__((ext_vector_type(2))) float v2f;
typedef __attribute__((ext_vector_type(8))) float v8f;

// --- CDNA5 async global->LDS path (guarded; falls back to sync staging) ----
#if defined(__gfx1250__) && __has_builtin(__builtin_amdgcn_global_load_async_to_lds_b128)
#define HAVE_ASYNC_LDS 1
typedef int v4i_t __attribute__((ext_vector_type(4)));
typedef __attribute__((address_space(1))) v4i_t* as1_v4p;   // global (AS1)
typedef __attribute__((address_space(3))) v4i_t* as3_v4p;   // LDS (AS3)
#else
#define HAVE_ASYNC_LDS 0
#endif

__device__ __forceinline__ void wait_async0() {
#if __has_builtin(__builtin_amdgcn_s_wait_asynccnt)
    __builtin_amdgcn_s_wait_asynccnt(0);
#else
    asm volatile("s_wait_asynccnt 0x0" ::: "memory");
#endif
}

__device__ __forceinline__ float fast_sqrtf(float x) {
#if __has_builtin(__builtin_amdgcn_sqrtf)
    return __builtin_amdgcn_sqrtf(x);   // raw v_sqrt_f32 (~1 ulp)
#else
    return sqrtf(x);
#endif
}

// Insert v into ascending sorted 5-list b[0..4] keeping the 5 smallest.
__device__ __forceinline__ void insert5(float (&b)[KTOP], float v) {
    if (v < b[4]) {
        float t = v, u;
        u = fmaxf(b[0], t); b[0] = fminf(b[0], t); t = u;
        u = fmaxf(b[1], t); b[1] = fminf(b[1], t); t = u;
        u = fmaxf(b[2], t); b[2] = fminf(b[2], t); t = u;
        u = fmaxf(b[3], t); b[3] = fminf(b[3], t); t = u;
        b[4] = fminf(b[4], t);
    }
}

// ---------------------------------------------------------------------------
// Kernel 1: per-row squared norms -> ws[inp*N + row]
// ---------------------------------------------------------------------------
__global__ void sqnorm_kernel(const float* __restrict__ x1,
                              const float* __restrict__ x2,
                              float* __restrict__ ws) {
    const int inp = blockIdx.y;
    const float* X = inp ? x2 : x1;
    const int row = blockIdx.x * blockDim.x + threadIdx.x;
    if (row >= N) return;
    const float4* xp = (const float4*)(X + (size_t)row * D);
    float s = 0.0f;
#pragma unroll 8
    for (int q = 0; q < D / 4; ++q) {
        float4 v = xp[q];
        s += v.x * v.x + v.y * v.y + v.z * v.z + v.w * v.w;
    }
    ws[inp * N + row] = s;
}

// ---------------------------------------------------------------------------
// Kernel 2: fused Gram-tile WMMA + streaming top-5-smallest per row.
//   grid = (N/128, 2), block = 256 (8 waves). Wave w owns rows
//   rblk + 16*w .. +15 and computes a 16x64 WMMA tile per panel iteration.
//   Panel of 64 columns is double-buffered in LDS; staging is async on the
//   CDNA5 path (GLOBAL_LOAD_ASYNC_TO_LDS_B128, ASYNCcnt) and overlaps the
//   WMMA K-loop of the current panel.
//   ws layout: [0,2N) row sqnorms, [2N,4N) per-row top5 sums.
// ---------------------------------------------------------------------------
__global__ void __launch_bounds__(256, 1)
dtm_topk_kernel(const float* __restrict__ x1,
                const float* __restrict__ x2,
                float* __restrict__ ws) {
    const int inp = blockIdx.y;
    const float* X   = inp ? x2 : x1;
    const float* sqn = ws + (size_t)inp * N;
    float*       S   = ws + 2 * N + (size_t)inp * N;

    __shared__ float ldsB[2 * PANEL * LDSB_STRIDE];   // 2 x 66,560 B
    float* const buf0 = ldsB;
    float* const buf1 = ldsB + PANEL * LDSB_STRIDE;

    const int tid   = threadIdx.x;
    const int wave  = tid >> 5;
    const int lane  = tid & 31;
    const int l15   = lane & 15;
    const int half8 = (lane >> 4) << 3;   // 0 or 8 : row-offset of my D-matrix half
    const int koff  = (lane >> 4) << 1;   // 0 or 2 : K-pair offset in A/B fragment

    const int rblk  = blockIdx.x * ROWS_PER_BLOCK;
    const int rbase = rblk + wave * 16;

    // A fragment base: lane holds row (rbase + l15), K pair (koff, koff+1)
    const float* arow = X + (size_t)(rbase + l15) * D + koff;

    // staging assignment: this thread owns column (tid>>2), k-chunk (tid&3)*64
    const int sc  = tid >> 2;
    const int sk0 = (tid & 3) * 64;

    float sqr_[8];
#pragma unroll
    for (int v = 0; v < 8; ++v) sqr_[v] = sqn[rbase + half8 + v];

    float best[8][KTOP];
#pragma unroll
    for (int v = 0; v < 8; ++v)
#pragma unroll
        for (int j = 0; j < KTOP; ++j) best[v][j] = 3.0e38f;

    const v8f vzero = {0.f, 0.f, 0.f, 0.f, 0.f, 0.f, 0.f, 0.f};

#if HAVE_ASYNC_LDS
    // prologue: async-stage panel 0 into buf0
    {
        const float* src = X + (size_t)(0 * PANEL + sc) * D + sk0;
        float* dst = &buf0[sc * LDSB_STRIDE + sk0];
#pragma unroll
        for (int q = 0; q < 16; ++q)
            __builtin_amdgcn_global_load_async_to_lds_b128(
                (as1_v4p)(src + q * 4), (as3_v4p)(dst + q * 4), 0, 0);
    }
#endif

    for (int p = 0; p < NPANELS; ++p) {
        const int cbase = p * PANEL;
        float* cur = (p & 1) ? buf1 : buf0;

#if HAVE_ASYNC_LDS
        wait_async0();          // my async writes of panel p are in LDS
        __syncthreads();        // everyone's are; prior reads of `nxt` done
        if (p + 1 < NPANELS) {  // prefetch next panel while we compute
            float* nxt = (p & 1) ? buf0 : buf1;
            const float* src = X + (size_t)((p + 1) * PANEL + sc) * D + sk0;
            float* dst = &nxt[sc * LDSB_STRIDE + sk0];
#pragma unroll
            for (int q = 0; q < 16; ++q)
                __builtin_amdgcn_global_load_async_to_lds_b128(
                    (as1_v4p)(src + q * 4), (as3_v4p)(dst + q * 4), 0, 0);
        }
#else
        {   // synchronous staging fallback
            const float4* src = (const float4*)(X + (size_t)(cbase + sc) * D + sk0);
            float4* dst = (float4*)(&cur[sc * LDSB_STRIDE + sk0]);
#pragma unroll
            for (int q = 0; q < 16; ++q) dst[q] = src[q];
        }
        __syncthreads();
#endif

        v8f acc[4] = {vzero, vzero, vzero, vzero};

        // --- K loop: 64 steps of V_WMMA_F32_16X16X4_F32, 4 tiles wide ---
#pragma unroll 4
        for (int ks = 0; ks < D / 4; ++ks) {
            const int k0 = ks * 4;
            v2f a = *(const v2f*)(arow + k0);
#pragma unroll
            for (int t = 0; t < 4; ++t) {
                v2f b = *(const v2f*)(&cur[(t * 16 + l15) * LDSB_STRIDE + k0 + koff]);
                acc[t] = __builtin_amdgcn_wmma_f32_16x16x4_f32(
                    false, a, false, b, (short)0, acc[t], false, false);
            }
        }

        // --- convert Gram to distance, stream into per-row top-5 lists ---
#pragma unroll
        for (int t = 0; t < 4; ++t) {
            const float scn = sqn[cbase + t * 16 + l15];
#pragma unroll
            for (int v = 0; v < 8; ++v) {
                float d2 = sqr_[v] + scn - 2.0f * acc[t][v];
                float dv = fast_sqrtf(fmaxf(d2, 0.0f));
                insert5(best[v], dv);
            }
        }
        __syncthreads();   // all waves done reading `cur` before it is restaged
    }

    // --- epilogue: merge 16 per-lane candidate lists per row via LDS ---
    float* cand = buf0;    // reuse: 128 rows * 81 floats = 10368 <= 16640
#pragma unroll
    for (int v = 0; v < 8; ++v) {
        const int rs = wave * 16 + half8 + v;   // row slot within block
#pragma unroll
        for (int j = 0; j < KTOP; ++j)
            cand[rs * 81 + l15 * KTOP + j] = best[v][j];
    }
    __syncthreads();

    if (tid < ROWS_PER_BLOCK) {
        float b5[KTOP];
#pragma unroll
        for (int j = 0; j < KTOP; ++j) b5[j] = 3.0e38f;
        const float* c = &cand[tid * 81];
        for (int i = 0; i < 16 * KTOP; ++i) insert5(b5, c[i]);
        S[rblk + tid] = b5[0] + b5[1] + b5[2] + b5[3] + b5[4];
    }
}

// ---------------------------------------------------------------------------
// Kernel 3: MSE over the two 8192-length sum vectors -> scalar
// ---------------------------------------------------------------------------
__global__ void finalize_kernel(const float* __restrict__ S, float* __restrict__ out) {
    __shared__ float red[256];
    const int tid = threadIdx.x;
    float a = 0.0f;
    for (int i = tid; i < N; i += 256) {
        float d = S[i] - S[N + i];
        a += d * d;
    }
    red[tid] = a;
    __syncthreads();
    for (int s = 128; s > 0; s >>= 1) {
        if (tid < s) red[tid] += red[tid + s];
        __syncthreads();
    }
    if (tid == 0) out[0] = red[0] / (float)N;
}

// ---------------------------------------------------------------------------
extern "C" void kernel_launch(void* const* d_in, const int* in_sizes, int n_in,
                              void* d_out, int out_size, void* d_ws, size_t ws_size,
                              hipStream_t stream) {
    const float* x1 = (const float*)d_in[0];
    const float* x2 = (const float*)d_in[1];
    float* ws  = (float*)d_ws;    // needs 4*N floats = 128 KB
    float* out = (float*)d_out;

    sqnorm_kernel<<<dim3(N / 256, 2), 256, 0, stream>>>(x1, x2, ws);
    dtm_topk_kernel<<<dim3(N / ROWS_PER_BLOCK, 2), 256, 0, stream>>>(x1, x2, ws);
    finalize_kernel<<<1, 256, 0, stream>>>(ws + 2 * N, out);
}